// SelfAttentionGen128_65481071404528
// MI455X (gfx1250) — compile-verified
//
#include <hip/hip_runtime.h>

// ---------------------------------------------------------------------------
// SelfAttention with per-token outer-product softmax (fp32).
//   Q = X @ Wr, K = X @ We  (2048x1024 @ 1024x1024 fp32 GEMMs) -> d_ws
//   out[b,s,h,i] = sum_j softmax_j(Q_i*K_j/8) * V_j,  V = X
//
// GEMM: V_WMMA_F32_16X16X4_F32 (the fp32 CDNA5 matrix op), double-buffered
// LDS staging via GLOBAL_LOAD_ASYNC_TO_LDS (ASYNCcnt) so DMA of chunk k+1
// overlaps the 32 WMMAs of chunk k. B is stored interleaved [k/2][n][k&1] so
// each lane's (k,k+1) fragment pair is a single aligned ds_load_b64, with the
// k-pair stride padded to 160 floats (160 mod 64 == 32) so the two half-wave
// row groups hit disjoint LDS bank halves.
// ---------------------------------------------------------------------------

typedef float v2f __attribute__((ext_vector_type(2)));
typedef float v8f __attribute__((ext_vector_type(8)));
typedef int   vec4i __attribute__((vector_size(16)));   // matches builtin sig

// Builtin parameter types (clang prints AS1 as __device__, AS3 as __shared__):
typedef __attribute__((address_space(1))) vec4i* gptr_b128;
typedef __attribute__((address_space(3))) vec4i* lptr_b128;
typedef __attribute__((address_space(1))) int*   gptr_b32;
typedef __attribute__((address_space(3))) int*   lptr_b32;

#define GM 2048   // rows of inputs (b*s)
#define GK 1024   // embed dim (reduction)
#define GN 1024   // output dim
#define NH 16
#define HD 64

#define TM 128    // block tile M (4 waves x 32 rows)
#define TN 64     // block tile N
#define TK 16     // K chunk (4 WMMA k-steps)
#define NCHUNK (GK / TK)
#define LDA 20    // padded A row stride: m*20 mod 64 distinct, float4 aligned
#define KPB (TK / 2)
#define LDBP 160  // interleaved-B k-pair stride: 160 mod 64 = 32 -> half-wave
                  // groups read disjoint bank halves, b64 conflict-free

#if __has_builtin(__builtin_amdgcn_global_load_async_to_lds_b128) && \
    __has_builtin(__builtin_amdgcn_global_load_async_to_lds_b32)  && \
    __has_builtin(__builtin_amdgcn_s_wait_asynccnt)
#define USE_ASYNC 1
#else
#define USE_ASYNC 0
#endif

__device__ __forceinline__ void stage_tiles(
    const float* __restrict__ X, const float* __restrict__ W,
    float* __restrict__ Abuf, float* __restrict__ Bbuf,
    int m0, int n0, int kc, int tid)
{
    // ---- A tile 128x16: 512 float4 chunks, 4 per thread ----
    #pragma unroll
    for (int j = 0; j < 4; ++j) {
        int c   = tid + 128 * j;
        int row = c >> 2;
        int c4  = c & 3;
        const float* g = X + (size_t)(m0 + row) * GK + kc + c4 * 4;
        float*       l = Abuf + row * LDA + c4 * 4;
#if USE_ASYNC
        __builtin_amdgcn_global_load_async_to_lds_b128(
            (gptr_b128)g, (lptr_b128)l, 0, 0);
#else
        *(float4*)l = *(const float4*)g;
#endif
    }
    // ---- B tile 16x64 -> interleaved [k/2][n][k&1]: 8 b32 per thread ----
    #pragma unroll
    for (int j = 0; j < 8; ++j) {
        int e = tid + 128 * j;          // element id within 16x64 tile
        int k = e >> 6;
        int n = e & 63;
        const float* g = W + (size_t)(kc + k) * GN + n0 + n;
        float*       l = Bbuf + (k >> 1) * LDBP + n * 2 + (k & 1);
#if USE_ASYNC
        __builtin_amdgcn_global_load_async_to_lds_b32(
            (gptr_b32)g, (lptr_b32)l, 0, 0);
#else
        *l = *g;
#endif
    }
}

__global__ __launch_bounds__(128) void qk_proj_kernel(
    const float* __restrict__ X,
    const float* __restrict__ Wr,
    const float* __restrict__ We,
    float* __restrict__ Qo,
    float* __restrict__ Ko)
{
    const float* __restrict__ W = (blockIdx.z == 0) ? Wr : We;
    float* __restrict__ Y       = (blockIdx.z == 0) ? Qo : Ko;

    const int n0   = blockIdx.x * TN;
    const int m0   = blockIdx.y * TM;
    const int tid  = threadIdx.x;
    const int lane = tid & 31;          // wave32
    const int wave = tid >> 5;          // 4 waves, each owns 32 M rows
    const int hi   = lane >> 4;         // half-wave select (K pairing, ISA 7.12.2)
    const int lo   = lane & 15;

    __shared__ __align__(16) float As[2][TM * LDA];   // 2 x 10 KB
    __shared__ __align__(16) float Bs[2][KPB * LDBP]; // 2 x 5 KB

    v8f zero = {};
    v8f acc[2][4];
    #pragma unroll
    for (int mt = 0; mt < 2; ++mt)
        #pragma unroll
        for (int nt = 0; nt < 4; ++nt) acc[mt][nt] = zero;

    stage_tiles(X, W, As[0], Bs[0], m0, n0, 0, tid);

    for (int ic = 0; ic < NCHUNK; ++ic) {
        const int cur = ic & 1;
#if USE_ASYNC
        __builtin_amdgcn_s_wait_asynccnt(0);   // own DMA done
#endif
        __syncthreads();                       // everyone's DMA/compute done

        // Kick off DMA of the next chunk into the other buffer; it streams
        // while the 32 WMMAs below execute.
        if (ic + 1 < NCHUNK)
            stage_tiles(X, W, As[cur ^ 1], Bs[cur ^ 1],
                        m0, n0, (ic + 1) * TK, tid);

        const float* __restrict__ Ab = As[cur];
        const float* __restrict__ Bb = Bs[cur];
        #pragma unroll
        for (int kk = 0; kk < 4; ++kk) {
            const int kb = kk * 4 + 2 * hi;    // this lane's K pair
            // A frags (16x4 f32): lane -> M=lo, VGPRs hold K=kb,kb+1
            v2f a0 = *(const v2f*)(Ab + (wave * 32 +      lo) * LDA + kb);
            v2f a1 = *(const v2f*)(Ab + (wave * 32 + 16 + lo) * LDA + kb);
            #pragma unroll
            for (int nt = 0; nt < 4; ++nt) {
                // B frag pair contiguous in LDS -> one ds_load_b64
                v2f b = *(const v2f*)(Bb + (kk * 2 + hi) * LDBP
                                         + (nt * 16 + lo) * 2);
                acc[0][nt] = __builtin_amdgcn_wmma_f32_16x16x4_f32(
                    false, a0, false, b, (short)0, acc[0][nt], false, false);
                acc[1][nt] = __builtin_amdgcn_wmma_f32_16x16x4_f32(
                    false, a1, false, b, (short)0, acc[1][nt], false, false);
            }
        }
    }

    // ---- write C: lane l, VGPR r -> M = r + 8*(l>>4), N = l&15 ----
    #pragma unroll
    for (int mt = 0; mt < 2; ++mt)
        #pragma unroll
        for (int nt = 0; nt < 4; ++nt)
            #pragma unroll
            for (int r = 0; r < 8; ++r) {
                int m = m0 + wave * 32 + mt * 16 + r + 8 * hi;
                int n = n0 + nt * 16 + lo;
                Y[(size_t)m * GN + n] = acc[mt][nt][r];
            }
}

// ---------------------------------------------------------------------------
// Attention: scores[i,j] = q_i*k_j/8 is rank-1, so softmax row i only needs
// k[], v[] and the scalar q_i.  One thread per output element; 4 token-heads
// per 256-thread block with k/v staged in LDS; v_exp_f32 for the softmax.
// ---------------------------------------------------------------------------
__global__ __launch_bounds__(256) void attn_kernel(
    const float* __restrict__ Q,
    const float* __restrict__ K,
    const float* __restrict__ V,
    float* __restrict__ O)
{
    const int tid = threadIdx.x;
    const int g   = tid >> 6;                    // token-head slot in block
    const int i   = tid & 63;                    // head_dim index
    const int th  = blockIdx.x * 4 + g;          // token-head id (0..32767)
    const size_t base = (size_t)(th >> 4) * GK + (size_t)(th & 15) * HD;

    __shared__ float ks[4][HD];
    __shared__ float vs[4][HD];

    ks[g][i] = K[base + i];
    vs[g][i] = V[base + i];
    const float qi = Q[base + i] * 0.125f;       // 1/sqrt(64)
    __syncthreads();

    float mx = -3.402823466e+38f;
    #pragma unroll 8
    for (int j = 0; j < HD; ++j) mx = fmaxf(mx, qi * ks[g][j]);

    float denom = 0.f, numer = 0.f;
    #pragma unroll 8
    for (int j = 0; j < HD; ++j) {
        float e = __expf(qi * ks[g][j] - mx);    // v_exp_f32
        denom += e;
        numer += e * vs[g][j];
    }
    O[base + i] = numer / denom;
}

extern "C" void kernel_launch(void* const* d_in, const int* in_sizes, int n_in,
                              void* d_out, int out_size, void* d_ws, size_t ws_size,
                              hipStream_t stream) {
    const float* rot = (const float*)d_in[0];   // rotation_params (1024,1024)
    const float* ent = (const float*)d_in[1];   // entangle_params (1024,1024)
    const float* x   = (const float*)d_in[2];   // inputs (2,1024,1024)
    float* out = (float*)d_out;

    float* Qw = (float*)d_ws;                   // 2048*1024 f32 = 8 MB
    float* Kw = Qw + (size_t)GM * GN;           // next 8 MB

    dim3 gridP(GN / TN, GM / TM, 2);            // 16 x 16 x {Q,K}
    qk_proj_kernel<<<gridP, 128, 0, stream>>>(x, rot, ent, Qw, Kw);

    attn_kernel<<<(GM * NH) / 4, 256, 0, stream>>>(Qw, Kw, x, out);
}